// Self_attention_12309376270362
// MI455X (gfx1250) — compile-verified
//
#include <hip/hip_runtime.h>
#include <hip/hip_bf16.h>
#include <math.h>

// Problem constants (match reference)
#define BB 32
#define SS 1024
#define DD 512
#define CC 5
#define EPSF 1e-7f

typedef float v2f __attribute__((ext_vector_type(2)));
typedef float v8f __attribute__((ext_vector_type(8)));

#define QSTR 514      // 514%64==2 -> conflict-free column reads; rows 8B-aligned
#define KROW 516      // 516%64==4 -> conflict-free; rows 16B-aligned for async B128
#define TSTR 68       // 68%64==4  -> conflict-free t-tile A-frag reads
#define PSTR 17       // partial-score tile row stride
#define KCH  64       // keys per double-buffered chunk
#define NCHUNK (SS / KCH)   // 16

// Dynamic-LDS partition (floats):
//   k_lds : 2*64*KROW = 66048  (258.0 KB)  double-buffered key chunks
//   q_lds : 16*QSTR   =  8224  ( 32.1 KB)  projected queries
//   t_lds : 16*TSTR   =  1088  (  4.3 KB)  final score tile exchange
//   ps_lds: 8*16*PSTR =  2176  (  8.5 KB)  K-half partial score tiles
//   rs    : 16                             row-sum normalizer
#define SMEM_FLOATS (2 * 64 * KROW + 16 * QSTR + 16 * TSTR + 8 * 16 * PSTR + 16)

// ---------------------------------------------------------------------------
// Kernel 0: per-(b,s) opinion gate + mask as f32 (tiny, bandwidth trivial)
// ---------------------------------------------------------------------------
__global__ __launch_bounds__(256) void prep_kernel(
    const float* __restrict__ gold_op,     // [B,S,C]
    const float* __restrict__ pred_op,     // [B,S,C]
    const float* __restrict__ gold_prob,   // [B,1]
    const unsigned char* __restrict__ mask,// [B,S] bool
    float* __restrict__ opw,               // [B,S]
    float* __restrict__ maskf)             // [B,S]
{
    int idx = blockIdx.x * blockDim.x + threadIdx.x;
    if (idx >= BB * SS) return;
    int b = idx / SS;
    float gp = gold_prob[b];
    float g  = gold_op[idx * CC + 1] + gold_op[idx * CC + 2];
    float p  = pred_op[idx * CC + 3] + pred_op[idx * CC + 4];
    opw[idx]   = gp * g + (1.0f - gp) * p;
    maskf[idx] = mask[idx] ? 1.0f : 0.0f;
}

// One 16-byte-per-lane async global->LDS copy (CDNA5 ASYNCcnt path).
// LDS byte address = low 32 bits of the flat shared pointer (HW truncates
// flat LDS addresses to addr[31:0]).
__device__ __forceinline__ void async_copy_b128(const float* gsrc, float* ldst) {
    unsigned int       la = (unsigned int)(unsigned long long)(uintptr_t)ldst;
    unsigned long long ga = (unsigned long long)(uintptr_t)gsrc;
    asm volatile("global_load_async_to_lds_b128 %0, %1, off"
                 :: "v"(la), "v"(ga) : "memory");
}

// Stage one 64-key x 512-float chunk (128 KB): wave w DMAs rows 8w..8w+7,
// 4 x B128 issues per row (16 B per lane). 32 asyncs/wave, under the 63 cap.
__device__ __forceinline__ void stage_chunk(const float* xb, float* kbuf,
                                            int keybase, int wave, int lane4) {
    for (int r = 0; r < 8; ++r) {
        const int row = wave * 8 + r;
        const float* gsrc = xb + (size_t)(keybase + row) * DD;
        float*       ldst = kbuf + row * KROW;
        #pragma unroll
        for (int c = 0; c < 4; ++c)
            async_copy_b128(gsrc + c * 128 + lane4, ldst + c * 128 + lane4);
    }
}

// ---------------------------------------------------------------------------
// Kernel 1: fused projection + attention, one WG = (batch b, 16-query tile),
// 8 waves. Key chunks are double-buffered in LDS via async DMA so data
// movement overlaps compute; all matmuls use full-precision
// V_WMMA_F32_16X16X4_F32 (workload is HBM-bound: ~86 GFLOP vs ~130 MB).
// ---------------------------------------------------------------------------
__global__ __launch_bounds__(256) void attn_kernel(
    const float* __restrict__ x,     // [B,S,D]
    const float* __restrict__ W,     // [D,D]
    const float* __restrict__ bias,  // [D]
    const float* __restrict__ opw,   // [B,S]
    const float* __restrict__ maskf, // [B,S]
    float* __restrict__ out)         // [B,S,D]
{
    extern __shared__ float smem[];
    float* k_lds  = smem;                      // [2][64][KROW]
    float* q_lds  = k_lds + 2 * 64 * KROW;     // [16][QSTR]
    float* t_lds  = q_lds + 16 * QSTR;         // [16][TSTR]
    float* ps_lds = t_lds + 16 * TSTR;         // [8][16][PSTR]
    float* rs_lds = ps_lds + 8 * 16 * PSTR;    // [16]

    const int tid  = (int)threadIdx.x;
    const int wave = tid >> 5;          // 0..7
    const int lane = tid & 31;
    const int lo   = lane & 15;         // N / M index within half-wave
    const int hi   = lane >> 4;         // 0/1: which half
    const int lane4 = lane * 4;

    const int mb = (int)blockIdx.x * 16;   // query-tile base row
    const int b  = (int)blockIdx.y;

    const float* xb = x + (size_t)b * SS * DD;

    if (tid < 16) rs_lds[tid] = 0.0f;

    // Chunk-0 DMA hides under the projection GEMM below.
    stage_chunk(xb, k_lds, 0, wave, lane4);

    // ---------------- Phase 1: q = x @ W + bias for this 16-row tile -------
    {
        const float* xrow = xb + (size_t)(mb + lo) * DD;  // A row for this lane
        for (int nt = 0; nt < 4; ++nt) {
            const int n0   = wave * 64 + nt * 16;
            const int ncol = n0 + lo;
            v8f acc;
            const float bv = bias[ncol];           // bias replicated down rows
            #pragma unroll
            for (int v = 0; v < 8; ++v) acc[v] = bv;

            for (int k0 = 0; k0 < DD; k0 += 4) {
                const int kk = k0 + 2 * hi;
                v2f a;  a[0]  = xrow[kk];
                        a[1]  = xrow[kk + 1];
                v2f bf; bf[0] = W[(size_t)kk * DD + ncol];
                        bf[1] = W[(size_t)(kk + 1) * DD + ncol];
                acc = __builtin_amdgcn_wmma_f32_16x16x4_f32(
                          false, a, false, bf, (short)0, acc, false, false);
            }
            #pragma unroll
            for (int v = 0; v < 8; ++v)
                q_lds[(v + 8 * hi) * QSTR + n0 + lo] = acc[v];
        }
    }
    __syncthreads();

    // ---------------- Phase 2: pipelined attention over 16 key chunks ------
    v8f oacc[4];
    #pragma unroll
    for (int nt = 0; nt < 4; ++nt)
        #pragma unroll
        for (int v = 0; v < 8; ++v) oacc[nt][v] = 0.0f;

    float rsum[8];
    #pragma unroll
    for (int v = 0; v < 8; ++v) rsum[v] = 0.0f;

    for (int kc = 0; kc < NCHUNK; ++kc) {
        const int par = kc & 1;
        float* kbuf = k_lds + par * (64 * KROW);

        // Retire this chunk's DMA (issued last iteration / prologue).
        asm volatile("s_wait_asynccnt 0x0" ::: "memory");
        __syncthreads();

        // Prefetch next chunk into the other buffer; overlaps all compute
        // below (its consumers wait at the next loop head).
        if (kc + 1 < NCHUNK)
            stage_chunk(xb, k_lds + (par ^ 1) * (64 * KROW),
                        (kc + 1) * KCH, wave, lane4);

        // ---- partial scores: tile (wave&3), K-half (wave>>2) -------------
        const int kt   = wave & 3;
        const int khal = wave >> 2;
        v8f s;
        #pragma unroll
        for (int v = 0; v < 8; ++v) s[v] = 0.0f;

        const float* qrow = q_lds + lo * QSTR;              // A row (query)
        const float* krow = kbuf + (kt * 16 + lo) * KROW;   // B row (key)
        const int kbeg = khal * 256;
        for (int k0 = kbeg; k0 < kbeg + 256; k0 += 4) {
            const int kk = k0 + 2 * hi;
            v2f a;  a[0]  = qrow[kk];
                    a[1]  = qrow[kk + 1];
            v2f bf; bf[0] = krow[kk];
                    bf[1] = krow[kk + 1];
            s = __builtin_amdgcn_wmma_f32_16x16x4_f32(
                    false, a, false, bf, (short)0, s, false, false);
        }
        #pragma unroll
        for (int v = 0; v < 8; ++v)
            ps_lds[wave * 16 * PSTR + (v + 8 * hi) * PSTR + lo] = s[v];
        __syncthreads();

        // ---- combine halves + elementwise (waves 0..3) -------------------
        if (wave < 4) {
            const int   n_g = kc * KCH + wave * 16 + lo;   // key index
            const float ow  = opw[b * SS + n_g];
            const float mw  = maskf[b * SS + n_g];
            #pragma unroll
            for (int v = 0; v < 8; ++v) {
                const float full = s[v] +
                    ps_lds[(wave + 4) * 16 * PSTR + (v + 8 * hi) * PSTR + lo];
                const int   m_g  = mb + v + 8 * hi;        // query index
                const float dist = fabsf((float)(m_g - n_g));
                const float loc  = (m_g == n_g) ? 0.0f : 1.0f / (dist + EPSF);
                const float wgt  = full * loc * ow;
                float t = __expf(tanhf(wgt));
                t = (m_g == n_g) ? 0.0f : t * mw;
                rsum[v] += t;                              // denominator term
                t_lds[(v + 8 * hi) * TSTR + wave * 16 + lo] = t;
            }
        }
        __syncthreads();

        // ---- PV: out[16, 64-slice] += t(16x64) @ v(64 x 64-slice) --------
        const float* trow = t_lds + lo * TSTR;             // A row (query m=lo)
        #pragma unroll
        for (int ktt = 0; ktt < 4; ++ktt) {
            #pragma unroll
            for (int ks = 0; ks < 16; ks += 4) {
                const int kk = ktt * 16 + ks + 2 * hi;     // key within chunk
                v2f a;  a[0] = trow[kk];
                        a[1] = trow[kk + 1];
                #pragma unroll
                for (int nt = 0; nt < 4; ++nt) {
                    const int dcol = wave * 64 + nt * 16 + lo;
                    v2f bf; bf[0] = kbuf[(size_t)kk * KROW + dcol];
                            bf[1] = kbuf[(size_t)(kk + 1) * KROW + dcol];
                    oacc[nt] = __builtin_amdgcn_wmma_f32_16x16x4_f32(
                                   false, a, false, bf, (short)0, oacc[nt],
                                   false, false);
                }
            }
        }
        __syncthreads();   // t_lds / ps_lds reused; kbuf re-DMA'd next+1 iter
    }

    // ---------------- row-sum reduction (wave32-safe; waves 4..7 hold 0) ---
    #pragma unroll
    for (int v = 0; v < 8; ++v) {
        float r = rsum[v];
        r += __shfl_xor(r, 1);
        r += __shfl_xor(r, 2);
        r += __shfl_xor(r, 4);
        r += __shfl_xor(r, 8);   // stays within each 16-lane half
        rsum[v] = r;
    }
    if (wave < 4 && lo == 0) {
        #pragma unroll
        for (int v = 0; v < 8; ++v)
            atomicAdd(&rs_lds[v + 8 * hi], rsum[v]);
    }
    __syncthreads();

    // ---------------- normalize + store ------------------------------------
    float scale[8];
    #pragma unroll
    for (int v = 0; v < 8; ++v)
        scale[v] = 1.0f / (rs_lds[v + 8 * hi] + EPSF);

    float* ob = out + ((size_t)b * SS + mb) * DD;
    #pragma unroll
    for (int nt = 0; nt < 4; ++nt) {
        const int dcol = wave * 64 + nt * 16 + lo;
        #pragma unroll
        for (int v = 0; v < 8; ++v) {
            const int m = v + 8 * hi;
            ob[(size_t)m * DD + dcol] = oacc[nt][v] * scale[v];
        }
    }
}

// ---------------------------------------------------------------------------
extern "C" void kernel_launch(void* const* d_in, const int* in_sizes, int n_in,
                              void* d_out, int out_size, void* d_ws, size_t ws_size,
                              hipStream_t stream) {
    (void)in_sizes; (void)n_in; (void)out_size; (void)ws_size;
    const float*         x        = (const float*)d_in[0];
    const float*         gold_op  = (const float*)d_in[1];
    const float*         pred_op  = (const float*)d_in[2];
    const float*         gprob    = (const float*)d_in[3];
    const unsigned char* mask     = (const unsigned char*)d_in[4];
    const float*         W        = (const float*)d_in[5];
    const float*         bias     = (const float*)d_in[6];
    float*               out      = (float*)d_out;

    float* opw   = (float*)d_ws;            // [B,S]
    float* maskf = opw + (size_t)BB * SS;   // [B,S]  (total 256 KB of ws)

    prep_kernel<<<(BB * SS + 255) / 256, 256, 0, stream>>>(
        gold_op, pred_op, gprob, mask, opw, maskf);

    attn_kernel<<<dim3(SS / 16, BB), 256, SMEM_FLOATS * sizeof(float), stream>>>(
        x, W, bias, opw, maskf, out);
}